// SNN1_21878563406061
// MI455X (gfx1250) — compile-verified
//
#include <hip/hip_runtime.h>

// SNN forward: out[b,n] = (1/50) * sum_t spike_t, with
//   fc = relu(x @ W^T), mem = mem*0.2*(1-spike) + fc, spike = mem > 0.5
//
// HBM-bound: x = 2.03 GB read once (floor ~87us @ 23.3 TB/s). GEMM is only
// 11.2 GFLOP -> use full-precision V_WMMA_F32_16X16X4_F32 (exact fp32, cheap
// issue) and fuse the temporal scan through LDS so fc never touches HBM.

typedef float v2f __attribute__((ext_vector_type(2)));
typedef float v8f __attribute__((ext_vector_type(8)));

#define NB      8                 // batches per workgroup
#define T_STEPS 50
#define N_IN    620
#define N_OUT   11
#define ROWS    (NB * T_STEPS)    // 400 flattened (b,t) rows per WG
#define TILES   (ROWS / 16)       // 25 M-tiles of 16 rows
#define KSTEPS  (N_IN / 4)        // 155 K-steps of 4

__global__ __launch_bounds__(256)
void snn_fused_kernel(const float* __restrict__ x,   // [B, T, 620]
                      const float* __restrict__ W,   // [11, 620]
                      float* __restrict__ out)       // [B, 11]
{
    __shared__ float sW[16 * N_IN];     // W zero-padded to 16 output rows
    __shared__ float sFC[ROWS * 16];    // relu(fc) tiles, [400][16]

    const int tid  = threadIdx.x;
    const int wave = tid >> 5;          // wave32
    const int lane = tid & 31;
    const int nf   = lane & 15;         // A: row-in-tile; B/D: column n
    const int kh   = lane >> 4;         // K half-select (WMMA f32 layout)

    // ---- stage padded weights into LDS (~39 KB, read 155x per tile) ----
    for (int idx = tid; idx < 16 * N_IN; idx += 256) {
        int n = idx / N_IN;
        int k = idx - n * N_IN;
        sW[idx] = (n < N_OUT) ? W[n * N_IN + k] : 0.0f;
    }
    __syncthreads();

    const size_t rowBase = (size_t)blockIdx.x * ROWS;

    // ---- GEMM phase: D[16x16] tiles via v_wmma_f32_16x16x4_f32 ----
    for (int tile = wave; tile < TILES; tile += 8) {
        const int lr = tile * 16 + nf;  // local row this lane feeds as A
        // A layout 16x4 f32: lane<16 holds K={0,1}, lane>=16 holds K={2,3}
        // -> one contiguous b64 per lane per step, +16B immediate per step.
        const float* px = x + (rowBase + (size_t)lr) * N_IN + 2 * kh;
        // B layout 4x16 f32: lane nf holds W[nf][k0], W[nf][k0+1]
        const float* pw = sW + nf * N_IN + 2 * kh;

        v8f c = {0.f, 0.f, 0.f, 0.f, 0.f, 0.f, 0.f, 0.f};
        #pragma unroll 5
        for (int s = 0; s < KSTEPS; ++s) {
            v2f a = *(const v2f*)(px + 4 * s);
            v2f b = *(const v2f*)(pw + 4 * s);
            c = __builtin_amdgcn_wmma_f32_16x16x4_f32(
                    /*neg_a=*/false, a, /*neg_b=*/false, b,
                    /*c_mod=*/(short)0, c, /*reuse_a=*/false, /*reuse_b=*/false);
        }

        // D layout: VGPR r -> row (tile*16 + r + 8*kh), col nf. Apply ReLU.
        union { v8f v; float f[8]; } u; u.v = c;
        #pragma unroll
        for (int r = 0; r < 8; ++r) {
            float vv = u.f[r];
            sFC[(tile * 16 + r + 8 * kh) * 16 + nf] = (vv > 0.f) ? vv : 0.f;
        }
    }
    __syncthreads();

    // ---- temporal LIF scan (sequential in t, per (batch, channel)) ----
    if (tid < NB * 16) {
        const int bl = tid >> 4;        // local batch 0..7
        const int n  = tid & 15;        // output channel (n<11 valid)
        float mem = 0.f, spike = 0.f, sum = 0.f;
        #pragma unroll 5
        for (int t = 0; t < T_STEPS; ++t) {
            float fc = sFC[(bl * T_STEPS + t) * 16 + n];
            mem   = mem * (0.2f * (1.0f - spike)) + fc;
            spike = (mem > 0.5f) ? 1.0f : 0.0f;
            sum  += spike;
        }
        if (n < N_OUT) {
            out[((size_t)blockIdx.x * NB + bl) * N_OUT + n] = sum * 0.02f;
        }
    }
}

extern "C" void kernel_launch(void* const* d_in, const int* in_sizes, int n_in,
                              void* d_out, int out_size, void* d_ws, size_t ws_size,
                              hipStream_t stream) {
    (void)n_in; (void)out_size; (void)d_ws; (void)ws_size;
    const float* x = (const float*)d_in[0];   // [B, 50, 620] fp32
    const float* W = (const float*)d_in[1];   // [11, 620]    fp32
    float* out = (float*)d_out;               // [B, 11]      fp32

    const int batch = in_sizes[0] / (T_STEPS * N_IN);   // 16384
    const int wgs   = batch / NB;                       // 2048 workgroups
    snn_fused_kernel<<<wgs, 256, 0, stream>>>(x, W, out);
}